// MXIntGELU_13683765805188
// MI455X (gfx1250) — compile-verified
//
#include <hip/hip_runtime.h>
#include <math.h>

// MXInt(8,8, blocks of 1x32) quantize -> fixed-point GELU emulation -> MXInt requantize.
// Bandwidth-bound streaming kernel for MI455X (gfx1250, wave32).
//
// Mapping: 8 contiguous elements per lane (2x b128) => one 32-element MX block
// == 4 consecutive lanes. Shared-exponent reduction: 2x __shfl_xor (masks 1,2).

typedef float v4f __attribute__((ext_vector_type(4)));

__device__ __forceinline__ float clamp_i8f(float x) {
    // v_med3_f32: clip(x, -128, 127) in one instruction
    return __builtin_amdgcn_fmed3f(x, -128.0f, 127.0f);
}

// e = clip(ceil(log2(max(v, 2^-128))), -128, 127), computed exactly via frexp:
// v = fr * 2^k with fr in [0.5, 1)  =>  ceil(log2 v) = k-1 if fr==0.5 else k.
__device__ __forceinline__ int mx_block_exp(float v) {
    float g = fmaxf(v, 0x1.0p-128f);
    int k;
    float fr = frexpf(g, &k);          // v_frexp_mant_f32 / v_frexp_exp_i32_f32
    int e = (fr == 0.5f) ? (k - 1) : k;
    return min(max(e, -128), 127);
}

// Exact-enough GELU: x * Phi(x), Phi via Abramowitz-Stegun 7.1.26 erf
// (abs err <= 1.5e-7 << the 1/64 rounding grid of the 8-bit hash output).
// ~16 instrs: v_rcp_f32 + v_exp_f32 (trans, co-execute) + FMAs; |x| is a free
// input modifier, sign restored with v_bfi (copysign).
__device__ __forceinline__ float gelu_fast(float x) {
    float z  = fabsf(x) * 0.70710678118654752440f;        // |x|/sqrt(2)
    float t  = __builtin_amdgcn_rcpf(fmaf(0.3275911f, z, 1.0f));
    float p  = fmaf(fmaf(fmaf(fmaf(1.061405429f, t, -1.453152027f),
                              t, 1.421413741f),
                         t, -0.284496736f),
                    t, 0.254829592f) * t;
    float ex = __builtin_amdgcn_exp2f((z * z) * -1.4426950408889634f); // e^{-z^2}
    float er = fmaf(-p, ex, 1.0f);                        // erf(z), z >= 0
    er = copysignf(er, x);                                // odd extension
    return x * fmaf(0.5f, er, 0.5f);                      // x * Phi(x)
}

__device__ __forceinline__ float vmax8(const v4f& a, const v4f& b) {
    float m0 = fmaxf(fmaxf(fabsf(a.x), fabsf(a.y)), fmaxf(fabsf(a.z), fabsf(a.w)));
    float m1 = fmaxf(fmaxf(fabsf(b.x), fabsf(b.y)), fmaxf(fabsf(b.z), fabsf(b.w)));
    return fmaxf(m0, m1);
}

__global__ void __launch_bounds__(256) mxint_gelu_kernel(
        const float* __restrict__ x, float* __restrict__ y, unsigned int nvec8) {
    const unsigned int stride = gridDim.x * blockDim.x;
    const v4f* __restrict__ xv = (const v4f*)x;
    v4f* __restrict__ yv = (v4f*)y;

    for (unsigned int i = blockIdx.x * blockDim.x + threadIdx.x; i < nvec8; i += stride) {
        // Prefetch next grid-stride tile (lowers to global_prefetch_b8 on gfx1250).
        if (i + stride < nvec8)
            __builtin_prefetch((const void*)(xv + 2 * (size_t)(i + stride)), 0, 0);

        // Two back-to-back 128-bit non-temporal streaming loads (32 B contiguous
        // per lane; single-use data, keep it out of L2).
        v4f va = __builtin_nontemporal_load(xv + 2 * (size_t)i);
        v4f vb = __builtin_nontemporal_load(xv + 2 * (size_t)i + 1);

        // ---- input MXInt: shared exponent over the 32-element block (4 lanes) ----
        float la = vmax8(va, vb);
        la = fmaxf(la, __shfl_xor(la, 1));
        la = fmaxf(la, __shfl_xor(la, 2));
        const int e1 = mx_block_exp(la);

        v4f qa, qb;
        #pragma unroll
        for (int j = 0; j < 8; ++j) {
            float xj = (j < 4) ? va[j] : vb[j - 4];
            // mantissa: clip(round(x * 2^(7-e)), -128, 127); dequant qx = m * 2^(e-7)
            float m  = clamp_i8f(rintf(ldexpf(xj, 7 - e1)));
            float qx = ldexpf(m, e1 - 7);

            // hash-range: in_width=8, frac=4 -> (-8.0, 7.9375); outside: relu
            float relu = fmaxf(qx, 0.0f);

            float gl = gelu_fast(qx);
            // fixed-point quantize: clip(round(g*64), -128, 127) / 64
            float qg = clamp_i8f(rintf(gl * 64.0f)) * 0.015625f;
            // rescale to block-mantissa grid: floor(qg * 2^(5-e)) * 2^(e-5)
            float mg = floorf(ldexpf(qg, 5 - e1));
            qg = ldexpf(mg, e1 - 5);

            bool rem = (qx > -8.0f) && (qx < 7.9375f);
            float qo = rem ? qg : relu;
            if (j < 4) qa[j] = qo; else qb[j - 4] = qo;
        }

        // ---- output MXInt requantization (fresh block exponent) ----
        float lb = vmax8(qa, qb);
        lb = fmaxf(lb, __shfl_xor(lb, 1));
        lb = fmaxf(lb, __shfl_xor(lb, 2));
        const int e2 = mx_block_exp(lb);

        #pragma unroll
        for (int j = 0; j < 4; ++j) {
            qa[j] = ldexpf(clamp_i8f(rintf(ldexpf(qa[j], 7 - e2))), e2 - 7);
            qb[j] = ldexpf(clamp_i8f(rintf(ldexpf(qb[j], 7 - e2))), e2 - 7);
        }

        // Two 128-bit non-temporal streaming stores.
        __builtin_nontemporal_store(qa, yv + 2 * (size_t)i);
        __builtin_nontemporal_store(qb, yv + 2 * (size_t)i + 1);
    }
}

extern "C" void kernel_launch(void* const* d_in, const int* in_sizes, int n_in,
                              void* d_out, int out_size, void* d_ws, size_t ws_size,
                              hipStream_t stream) {
    (void)n_in; (void)out_size; (void)d_ws; (void)ws_size;
    const float* x = (const float*)d_in[0];
    float* y = (float*)d_out;

    const unsigned int n = (unsigned int)in_sizes[0];   // 2*4096*4096
    const unsigned int nvec8 = n / 8;                   // 8 floats (32 B) per lane-iter

    const int threads = 256;                            // 8 waves per workgroup
    int blocks = 4096;                                  // grid-stride; saturates all WGPs
    const unsigned int need = (nvec8 + threads - 1) / threads;
    if ((unsigned int)blocks > need) blocks = (int)need;
    if (blocks < 1) blocks = 1;

    mxint_gelu_kernel<<<blocks, threads, 0, stream>>>(x, y, nvec8);
}